// GraphAttention_gather_66262755442759
// MI455X (gfx1250) — compile-verified
//
#include <hip/hip_runtime.h>
#include <hip/hip_bf16.h>
#include <math.h>
#include <stdint.h>

// Problem constants (reference: B,N,D,H = 4,512,64,64)
#define GB 4
#define GN 512
#define GD 64
#define GH 64
#define SROW 513   // padded score-row stride (odd -> conflict-free A-fragment reads)

typedef __attribute__((ext_vector_type(2))) float v2f;
typedef __attribute__((ext_vector_type(8))) float v8f;

// Dynamic LDS layout (floats):
//   s_att   [16 * SROW]      scores -> attention weights      (32832 B)
//   s_nodes [512 * 64]       async-staged nodes[b]            (131072 B)
//   s_part  [4 * 16 * 64]    K-split partial sums             (16384 B)
#define LDS_ATT_F   (16 * SROW)
#define LDS_NODES_F (GN * GD)
#define LDS_PART_F  (4 * 16 * GD)
#define LDS_BYTES   ((LDS_ATT_F + LDS_NODES_F + LDS_PART_F) * 4)

// ---------------------------------------------------------------------------
// Kernel 1: U = nodes @ (W1+W3) + bias ; V = nodes @ (W2-W3)
// Grid: B*N/16 = 128 blocks. Block: 128 threads = 4 waves; wave w owns H-tile w.
// ---------------------------------------------------------------------------
__global__ __launch_bounds__(128) void gat_prep_uv(
    const float* __restrict__ nodes,   // (B,N,D)
    const float* __restrict__ W,       // (3D,H) = (192,64)
    const float* __restrict__ bias,    // (H)
    float* __restrict__ U,             // (B,N,H)
    float* __restrict__ V)             // (B,N,H)
{
    const int tile = blockIdx.x;           // 0..127
    const int b    = tile >> 5;            // 32 tiles per batch
    const int n0   = (tile & 31) << 4;     // row tile base
    const int wave = threadIdx.x >> 5;     // 0..3 -> H tile
    const int lane = threadIdx.x & 31;
    const int lr   = lane & 15;
    const int hi   = lane >> 4;            // 0: K pair {0,1}, 1: K pair {2,3}
    const int h    = wave * 16 + lr;       // output column

    const float* nb = nodes + (size_t)b * GN * GD + (size_t)n0 * GD;

    const float bv = bias[h];
    v8f cu, cv;
#pragma unroll
    for (int j = 0; j < 8; ++j) { cu[j] = bv; cv[j] = 0.0f; }

#pragma unroll
    for (int k0 = 0; k0 < GD; k0 += 4) {
        // A fragment (16x4 of nodes): lane row M=lr, K = k0 + 2*hi + {0,1}
        const float* ap = nb + lr * GD + k0 + (hi << 1);
        v2f a; a.x = ap[0]; a.y = ap[1];

        // B fragments (4x16): VGPR0 -> K row (k0+2*hi), VGPR1 -> next K row, col h
        const int kr = k0 + (hi << 1);
        const float w1a = W[(kr + 0) * GH + h];
        const float w1b = W[(kr + 1) * GH + h];
        const float w2a = W[(64 + kr + 0) * GH + h];
        const float w2b = W[(64 + kr + 1) * GH + h];
        const float w3a = W[(128 + kr + 0) * GH + h];
        const float w3b = W[(128 + kr + 1) * GH + h];
        v2f b13; b13.x = w1a + w3a; b13.y = w1b + w3b;
        v2f b23; b23.x = w2a - w3a; b23.y = w2b - w3b;

        cu = __builtin_amdgcn_wmma_f32_16x16x4_f32(false, a, false, b13,
                                                   (short)0, cu, false, false);
        cv = __builtin_amdgcn_wmma_f32_16x16x4_f32(false, a, false, b23,
                                                   (short)0, cv, false, false);
    }

    // C/D layout: VGPR j -> row M = j + 8*hi, col = h
    float* up = U + (size_t)b * GN * GH + (size_t)n0 * GH;
    float* vp = V + (size_t)b * GN * GH + (size_t)n0 * GH;
#pragma unroll
    for (int j = 0; j < 8; ++j) {
        const int M = j + (hi << 3);
        up[M * GH + h] = cu[j];
        vp[M * GH + h] = cv[j];
    }
}

// ---------------------------------------------------------------------------
// Kernel 2: edge scores -> softmax -> messgs = att @ nodes
// Grid: B*N/16 = 128 blocks. Block: 512 threads = 16 waves.
// nodes[b] is async-staged into LDS (overlapped with phases 1-2).
// ---------------------------------------------------------------------------
__global__ __launch_bounds__(512) void gat_attn(
    const float* __restrict__ nodes,   // (B,N,D)
    const int*   __restrict__ adjs,    // (N,N,2)
    const float* __restrict__ avec,    // (H)
    const float* __restrict__ U,       // (B,N,H)  (bias folded in)
    const float* __restrict__ V,       // (B,N,H)
    float* __restrict__ out)           // (B,N,D)
{
    extern __shared__ float smem[];
    float* s_att   = smem;
    float* s_nodes = smem + LDS_ATT_F;
    float* s_part  = smem + LDS_ATT_F + LDS_NODES_F;

    const int blk  = blockIdx.x;
    const int b    = blk >> 5;
    const int n0   = (blk & 31) << 4;
    const int tid  = threadIdx.x;
    const int lane = tid & 31;

    // ---- Phase 0: kick off async copy of nodes[b] (128KB) into LDS --------
    {
        const float* nb = nodes + (size_t)b * GN * GD;     // wave-uniform base
        const uint32_t ldsbase = (uint32_t)(uintptr_t)s_nodes;
#pragma unroll
        for (int it = 0; it < 16; ++it) {
            const uint32_t byteoff = (uint32_t)(it * 512 + tid) * 16u;
            const uint32_t ldsaddr = ldsbase + byteoff;
            asm volatile("global_load_async_to_lds_b128 %0, %1, %2"
                         :: "v"(ldsaddr), "v"(byteoff), "s"(nb)
                         : "memory");
        }
    }

    const float* Ub = U + (size_t)b * GN * GH;
    const float* Vb = V + (size_t)b * GN * GH;
    const float4* aa = (const float4*)avec;

    // ---- Phase 1: scores[r][m] = norm * sum_h a_h * leaky(U[i]+V[j]) -------
    const float norm = 0.125f;  // 1/sqrt(D)
#pragma unroll 1
    for (int p = tid; p < 16 * GN; p += 512) {
        const int r = p >> 9;          // row in tile
        const int m = p & (GN - 1);
        const int n = n0 + r;
        const int eb = ((n << 9) + m) << 1;
        __builtin_prefetch(&adjs[eb + 1024], 0, 1);
        const int ii = adjs[eb + 0];
        const int jj = adjs[eb + 1];
        const float4* uu = (const float4*)(Ub + (size_t)ii * GH);
        const float4* vv = (const float4*)(Vb + (size_t)jj * GH);
        float acc = 0.0f;
#pragma unroll
        for (int q = 0; q < GH / 4; ++q) {
            const float4 u4 = uu[q];
            const float4 v4 = vv[q];
            const float4 a4 = aa[q];
            float x;
            x = u4.x + v4.x; acc += a4.x * (x > 0.0f ? x : 0.2f * x);
            x = u4.y + v4.y; acc += a4.y * (x > 0.0f ? x : 0.2f * x);
            x = u4.z + v4.z; acc += a4.z * (x > 0.0f ? x : 0.2f * x);
            x = u4.w + v4.w; acc += a4.w * (x > 0.0f ? x : 0.2f * x);
        }
        s_att[r * SROW + m] = acc * norm;
    }
    __syncthreads();

    // ---- Phase 2: softmax over m, one wave32 per row -----------------------
    {
        const int w = tid >> 5;        // 16 waves -> 16 rows
        float* row = s_att + w * SROW;

        float mx = -INFINITY;
        for (int m = lane; m < GN; m += 32) mx = fmaxf(mx, row[m]);
#pragma unroll
        for (int off = 16; off > 0; off >>= 1)
            mx = fmaxf(mx, __shfl_xor(mx, off, 32));

        float sum = 0.0f;
        for (int m = lane; m < GN; m += 32) {
            const float e = __expf(row[m] - mx);
            row[m] = e;
            sum += e;
        }
#pragma unroll
        for (int off = 16; off > 0; off >>= 1)
            sum += __shfl_xor(sum, off, 32);

        const float inv = 1.0f / sum;
        for (int m = lane; m < GN; m += 32) row[m] *= inv;
    }

    // staged nodes must have landed before phase 3; fence ASYNCcnt, then sync
    asm volatile("s_wait_asynccnt 0" ::: "memory");
    __syncthreads();

    // ---- Phase 3: messgs(16x64) = att(16x512) @ nodes(512x64) --------------
    // 16 waves: wave q -> (K-chunk kq = q>>2 of 128, col-tile w = q&3 of 16)
    {
        const int q   = tid >> 5;
        const int kq  = q >> 2;
        const int w   = q & 3;
        const int lr  = lane & 15;
        const int hi  = lane >> 4;
        const int col = w * 16 + lr;
        const int kb  = kq * 128;

        v8f c;
#pragma unroll
        for (int j = 0; j < 8; ++j) c[j] = 0.0f;

#pragma unroll 4
        for (int k0 = 0; k0 < 128; k0 += 4) {
            const int kr = kb + k0 + (hi << 1);
            v2f a;                       // A: att rows (LDS, padded stride)
            a.x = s_att[lr * SROW + kr + 0];
            a.y = s_att[lr * SROW + kr + 1];
            v2f bb;                      // B: staged nodes rows (LDS)
            bb.x = s_nodes[(kr + 0) * GD + col];
            bb.y = s_nodes[(kr + 1) * GD + col];
            c = __builtin_amdgcn_wmma_f32_16x16x4_f32(false, a, false, bb,
                                                      (short)0, c, false, false);
        }

#pragma unroll
        for (int j = 0; j < 8; ++j) {
            const int M = j + (hi << 3);
            s_part[kq * (16 * GD) + M * GD + col] = c[j];
        }
    }
    __syncthreads();

    // ---- Phase 4: reduce 4 K-chunk partials, write output ------------------
    {
        float* ob = out + (size_t)b * GN * GD + (size_t)n0 * GD;
#pragma unroll
        for (int e = tid; e < 16 * GD; e += 512) {
            const float s = s_part[e]
                          + s_part[1 * 16 * GD + e]
                          + s_part[2 * 16 * GD + e]
                          + s_part[3 * 16 * GD + e];
            ob[e] = s;
        }
    }
}

// ---------------------------------------------------------------------------
extern "C" void kernel_launch(void* const* d_in, const int* in_sizes, int n_in,
                              void* d_out, int out_size, void* d_ws, size_t ws_size,
                              hipStream_t stream) {
    const float* nodes = (const float*)d_in[0];   // B*N*D
    const int*   adjs  = (const int*)d_in[1];     // N*N*2
    const float* W     = (const float*)d_in[2];   // 3D*H
    const float* bias  = (const float*)d_in[3];   // H
    const float* avec  = (const float*)d_in[4];   // H
    float* out = (float*)d_out;                   // B*N*D

    float* U = (float*)d_ws;                      // B*N*H floats
    float* V = U + (size_t)GB * GN * GH;          // B*N*H floats

    gat_prep_uv<<<GB * (GN / 16), 128, 0, stream>>>(nodes, W, bias, U, V);
    gat_attn<<<GB * (GN / 16), 512, LDS_BYTES, stream>>>(nodes, adjs, avec, U, V, out);
}